// FRNOD_18880676233811
// MI455X (gfx1250) — compile-verified
//
#include <hip/hip_runtime.h>
#include <cstdint>

#define MAXG 256
#define POS_T 0.7f
#define NEG_T 0.3f
#define NSAMP 256
#define NPOS  128

// ---- CDNA5 async global->LDS staging (ASYNCcnt-tracked) -------------------
#if defined(__has_builtin)
#if __has_builtin(__builtin_amdgcn_global_load_async_to_lds_b128)
#define USE_ASYNC_LDS 1
#endif
#if __has_builtin(__builtin_amdgcn_s_wait_asynccnt)
#define USE_WAIT_ASYNC_BUILTIN 1
#endif
#endif

#ifdef USE_ASYNC_LDS
typedef int v4i_b128 __attribute__((vector_size(16)));
typedef __attribute__((address_space(1))) v4i_b128* gbl_v4i_ptr;
typedef __attribute__((address_space(3))) v4i_b128* lds_v4i_ptr;
#endif

__device__ __forceinline__ void async_copy_b128(const void* gsrc, void* lds_dst) {
#ifdef USE_ASYNC_LDS
  __builtin_amdgcn_global_load_async_to_lds_b128(
      (gbl_v4i_ptr)(uintptr_t)gsrc,
      (lds_v4i_ptr)(uintptr_t)lds_dst,
      0, 0);
#else
  *(float4*)lds_dst = *(const float4*)gsrc;
#endif
}

__device__ __forceinline__ void wait_async_copies() {
#ifdef USE_ASYNC_LDS
#ifdef USE_WAIT_ASYNC_BUILTIN
  __builtin_amdgcn_s_wait_asynccnt(0);
#else
  asm volatile("s_wait_asynccnt 0x0" ::: "memory");
#endif
#endif
}

// ---- deterministic per-index "uniform" priority ---------------------------
__device__ __forceinline__ unsigned hash_u32(unsigned i, unsigned seed) {
  unsigned h = i ^ seed;
  h ^= h >> 16; h *= 0x7feb352du;
  h ^= h >> 15; h *= 0x846ca68bu;
  h ^= h >> 16;
  return h;
}

// ---- kernel 1: init workspace --------------------------------------------
__global__ void atc_init(unsigned long long* gbest, int* counters, int G) {
  int i = blockIdx.x * blockDim.x + threadIdx.x;
  if (i < G) gbest[i] = 0ull;
  if (i < 8) counters[i] = 0;
}

// ---- kernel 2: IoU sweep, row argmax, column max via packed-key atomics ---
__global__ void atc_iou(const float4* __restrict__ anchor,
                        const float4* __restrict__ bbox,
                        int K, int G,
                        int* __restrict__ argmax,
                        int* __restrict__ label,
                        unsigned long long* __restrict__ gbest) {
  __shared__ float4 sbox[MAXG];
  __shared__ float sareab[MAXG];
  __shared__ unsigned long long sbest[MAXG];
  const int tid = threadIdx.x;

  for (int g = tid; g < G; g += blockDim.x) {
    async_copy_b128(&bbox[g], &sbox[g]);
    sbest[g] = 0ull;
  }
  wait_async_copies();
  __syncthreads();
  for (int g = tid; g < G; g += blockDim.x) {
    float4 b = sbox[g];
    sareab[g] = (b.z - b.x) * (b.w - b.y);
  }
  __syncthreads();

  for (int i = blockIdx.x * blockDim.x + tid; i < K; i += gridDim.x * blockDim.x) {
    float4 a = anchor[i];
    float area_a = (a.z - a.x) * (a.w - a.y);
    float best = -1.0f;
    int arg = 0;
    for (int g = 0; g < G; ++g) {
      float4 b = sbox[g];
      float tlx = fmaxf(a.x, b.x), tly = fmaxf(a.y, b.y);
      float brx = fminf(a.z, b.z), bry = fminf(a.w, b.w);
      float inter = (tlx < brx && tly < bry) ? (brx - tlx) * (bry - tly) : 0.0f;
      float iou = inter / (area_a + sareab[g] - inter);
      if (iou > best) { best = iou; arg = g; }   // strict > keeps first max (argmax tie rule)
      // monotone packed key: iou bits high (iou >= 0), ~idx low -> ties pick lowest idx
      unsigned long long key =
          ((unsigned long long)__float_as_uint(iou) << 32) | (unsigned)~(unsigned)i;
      if (key > sbest[g]) atomicMax(&sbest[g], key);   // ds_max_u64
    }
    argmax[i] = arg;
    int lab = -1;
    if (best < NEG_T) lab = 0;
    if (best >= POS_T) lab = 1;
    label[i] = lab;
  }
  __syncthreads();
  for (int g = tid; g < G; g += blockDim.x)
    atomicMax(&gbest[g], sbest[g]);                    // global_atomic_max_u64
}

// ---- kernel 3: force each gt's best anchor (sequential -> last-wins) ------
__global__ void atc_force(const unsigned long long* __restrict__ gbest,
                          int* __restrict__ label, int* __restrict__ argmax,
                          int G, int K) {
  if (threadIdx.x == 0 && blockIdx.x == 0) {
    for (int g = 0; g < G; ++g) {
      unsigned idx = ~(unsigned)(gbest[g] & 0xFFFFFFFFull);
      if (idx < (unsigned)K) { label[idx] = 1; argmax[idx] = g; }
    }
  }
}

__device__ __forceinline__ int block_reduce_sum_256(int v, int* red) {
  const int tid = threadIdx.x;
  red[tid] = v;
  __syncthreads();
  for (int s = 128; s > 0; s >>= 1) {
    if (tid < s) red[tid] += red[tid + s];
    __syncthreads();
  }
  int r = red[0];
  __syncthreads();
  return r;
}

// ---- kernels 4/5: subsample labels (radix-select n-th smallest priority) --
// which==0: positives (target label 1, n=128, writes counters[1]=kept)
// which==1: negatives (target label 0, n=256-counters[1])
__global__ void atc_subsample(int* __restrict__ label, int* __restrict__ counters,
                              int K, int which) {
  __shared__ int red[256];
  const int tid = threadIdx.x;
  const int target = (which == 0) ? 1 : 0;
  const unsigned seed = (which == 0) ? 0x9E3779B9u : 0x85EBCA6Bu;

  int local = 0;
  for (int i = tid; i < K; i += 256)
    local += (label[i] == target);
  int P = block_reduce_sum_256(local, red);

  int n = (which == 0) ? NPOS : (NSAMP - counters[1]);
  if (n < 0) n = 0;

  if (P <= n) {                       // keep all
    if (which == 0 && tid == 0) counters[1] = P;
    return;
  }
  if (which == 0 && tid == 0) counters[1] = n;

  // radix select the (n-1)-th smallest hash key among masked elements
  unsigned prefix = 0;
  int k = n - 1;
  for (int b = 31; b >= 0; --b) {
    unsigned lo = prefix;
    unsigned hi = prefix | ((1u << b) - 1u);   // keys with bit b == 0 under prefix
    int c = 0;
    for (int i = tid; i < K; i += 256) {
      if (label[i] == target) {
        unsigned key = hash_u32((unsigned)i, seed);
        c += (key >= lo && key <= hi);
      }
    }
    c = block_reduce_sum_256(c, red);
    if (k >= c) { k -= c; prefix |= (1u << b); }
  }
  const unsigned Tk = prefix;

  int cl = 0;
  for (int i = tid; i < K; i += 256)
    if (label[i] == target && hash_u32((unsigned)i, seed) < Tk) cl++;
  cl = block_reduce_sum_256(cl, red);

  for (int i = tid; i < K; i += 256)
    if (label[i] == target && hash_u32((unsigned)i, seed) > Tk) label[i] = -1;
  __syncthreads();

  if (tid == 0) {                     // resolve ties at Tk by index (stable-sort rule)
    int r = n - cl, seen = 0;
    for (int i = 0; i < K; ++i) {
      if (label[i] == target && hash_u32((unsigned)i, seed) == Tk) {
        if (seen >= r) label[i] = -1; else seen++;
      }
    }
  }
}

// ---- kernel 6: bbox2loc + emit outputs ------------------------------------
__global__ void atc_finalize(const float4* __restrict__ anchor,
                             const float4* __restrict__ bbox,
                             const int* __restrict__ argmax,
                             const int* __restrict__ label,
                             const int* __restrict__ counters,
                             float* __restrict__ out, int K, int G) {
  __shared__ float4 sbox[MAXG];
  const int tid = threadIdx.x;
  for (int g = tid; g < G; g += blockDim.x)
    async_copy_b128(&bbox[g], &sbox[g]);
  wait_async_copies();
  __syncthreads();

  const bool has_pos = counters[1] > 0;
  const float EPS = 1.1920928955078125e-07f;  // FLT_EPSILON
  float4* locv = (float4*)out;
  float* labf = out + (size_t)4 * K;

  for (int i = blockIdx.x * blockDim.x + tid; i < K; i += gridDim.x * blockDim.x) {
    float4 a = anchor[i];
    int g = argmax[i];
    g = (g < 0) ? 0 : (g >= G ? G - 1 : g);
    float4 b = sbox[g];
    float aw = a.z - a.x, ah = a.w - a.y;
    float w = fmaxf(aw, EPS), h = fmaxf(ah, EPS);
    float cx = a.x + 0.5f * aw, cy = a.y + 0.5f * ah;
    float bw = b.z - b.x, bh = b.w - b.y;
    float bcx = b.x + 0.5f * bw, bcy = b.y + 0.5f * bh;
    float4 loc;
    if (has_pos) {
      loc.x = (bcx - cx) / w;
      loc.y = (bcy - cy) / h;
      loc.z = logf(bw / w);
      loc.w = logf(bh / h);
    } else {
      loc.x = 0.0f; loc.y = 0.0f; loc.z = 0.0f; loc.w = 0.0f;
    }
    locv[i] = loc;
    labf[i] = (float)label[i];
  }
}

extern "C" void kernel_launch(void* const* d_in, const int* in_sizes, int n_in,
                              void* d_out, int out_size, void* d_ws, size_t ws_size,
                              hipStream_t stream) {
  const float4* anchor = (const float4*)d_in[0];
  const float4* bbox   = (const float4*)d_in[1];
  int K = in_sizes[0] / 4;
  int G = in_sizes[1] / 4;
  if (G > MAXG) G = MAXG;
  float* out = (float*)d_out;

  // workspace carve-out (256B aligned slices)
  char* ws = (char*)d_ws;
  size_t off = 0;
  auto carve = [&](size_t bytes) -> void* {
    void* p = ws + off;
    off = (off + bytes + 255) & ~(size_t)255;
    return p;
  };
  unsigned long long* gbest = (unsigned long long*)carve((size_t)G * 8);
  int* counters = (int*)carve(8 * sizeof(int));
  int* argmax   = (int*)carve((size_t)K * sizeof(int));
  int* label    = (int*)carve((size_t)K * sizeof(int));

  const int threads = 256;                       // 8 wave32 waves per block
  const int blocks = (K + threads - 1) / threads;

  atc_init<<<(G + 255) / 256, 256, 0, stream>>>(gbest, counters, G);
  atc_iou<<<blocks, threads, 0, stream>>>(anchor, bbox, K, G, argmax, label, gbest);
  atc_force<<<1, 32, 0, stream>>>(gbest, label, argmax, G, K);
  atc_subsample<<<1, 256, 0, stream>>>(label, counters, K, /*which=*/0);
  atc_subsample<<<1, 256, 0, stream>>>(label, counters, K, /*which=*/1);
  atc_finalize<<<blocks, threads, 0, stream>>>(anchor, bbox, argmax, label, counters,
                                               out, K, G);
}